// HazardRNN_20478404067981
// MI455X (gfx1250) — compile-verified
//
#include <hip/hip_runtime.h>
#include <math.h>

typedef __attribute__((ext_vector_type(2))) float v2f;
typedef __attribute__((ext_vector_type(8))) float v8f;

#define N_SEQ 256
#define S_LEN 1024
#define HID   800
#define CHUNK 256

// Hardware tanh if the builtin exists on this toolchain (gfx1250 has V_TANH_F32).
__device__ __forceinline__ float fast_tanhf(float v) {
#if __has_builtin(__builtin_amdgcn_tanhf)
    return __builtin_amdgcn_tanhf(v);
#elif __has_builtin(__builtin_amdgcn_tanh_f32)
    return __builtin_amdgcn_tanh_f32(v);
#else
    return tanhf(v);
#endif
}

// ---------------------------------------------------------------------------
// Kernel 1: the recurrent scan.  h_j(t) = tanh(x_t * w_j + b_j + h_j(t-1))
// One block per sequence n; 256 threads (8 wave32); each thread owns 3-4
// hidden-unit chains for ILP across the serial tanh dependency.
// x[n,:] is staged into LDS in 256-float chunks with the CDNA5 async
// global->LDS datapath (GLOBAL_LOAD_ASYNC_TO_LDS_B32, ASYNCcnt).
// ---------------------------------------------------------------------------
__global__ __launch_bounds__(256) void rnn_scan_kernel(
    const float* __restrict__ x,      // [256, 1024]
    const float* __restrict__ W_in,   // [1, 800] -> [800]
    const float* __restrict__ b_in,   // [800]
    float* __restrict__ hbuf)         // [256, 800] workspace
{
    __shared__ float xs[CHUNK];
    const int n   = blockIdx.x;
    const int tid = threadIdx.x;

    float w[4], bb[4], h[4];
    int js[4];
    int nj = 0;
#pragma unroll
    for (int k = 0; k < 4; ++k) {
        int j = tid + k * 256;
        if (j < HID) {
            js[nj] = j;
            w[nj]  = W_in[j];
            bb[nj] = b_in[j];
            h[nj]  = 0.0f;
            ++nj;
        }
    }
    // nj == 4 exactly for tid < 32 (one full wave), else 3 -> no intra-wave
    // divergence in the hot loop.

    // Generic LDS pointer: low 32 bits == raw LDS byte offset on AMDGPU.
    const unsigned lds_dst = (unsigned)(uintptr_t)(&xs[tid]);
    const float* xr = x + (size_t)n * S_LEN;

    for (int c = 0; c < S_LEN / CHUNK; ++c) {
        __syncthreads();   // previous chunk fully consumed before overwrite
        {
            const float* gsrc = xr + c * CHUNK + tid;
            // copies 4 bytes per lane from global gsrc to the LDS address in lds_dst, tracked by ASYNCcnt.
            asm volatile("global_load_async_to_lds_b32 %0, %1, off"
                         :
                         : "v"(lds_dst), "v"(gsrc)
                         : "memory");
            asm volatile("s_wait_asynccnt 0" ::: "memory");
        }
        __syncthreads();
        for (int t = 0; t < CHUNK; ++t) {
            float xv = xs[t];
#pragma unroll
            for (int k = 0; k < 4; ++k) {
                if (k < nj) {
                    h[k] = fast_tanhf(fmaf(xv, w[k], bb[k] + h[k]));
                }
            }
        }
    }

    for (int k = 0; k < nj; ++k) {
        hbuf[(size_t)n * HID + js[k]] = h[k];
    }
}

// ---------------------------------------------------------------------------
// Kernel 2: out = softmax(h @ W_out + b_out) via V_WMMA_F32_16X16X4_F32.
// One wave32 per 16 rows of h (16 blocks).  N padded 2 -> 16 with zero
// columns in B (mask multiply, NOT a divergent load -> EXEC stays all-1s).
// K = 800 in steps of 4 -> 200 WMMA ops per wave.
//
// VGPR layouts (ISA 7.12.2):
//   A 16x4:  lanes 0-15 -> M=lane,   v0=K0,v1=K1; lanes 16-31 -> v0=K2,v1=K3
//   B 4x16:  lanes 0-15 -> N=lane,   v0=K0,v1=K1; lanes 16-31 -> v0=K2,v1=K3
//   C 16x16: VGPR r, lanes 0-15 -> M=r, N=lane; lanes 16-31 -> M=8+r
// ---------------------------------------------------------------------------
__global__ __launch_bounds__(32) void out_proj_softmax_kernel(
    const float* __restrict__ hbuf,   // [256, 800]
    const float* __restrict__ W_out,  // [800, 2]
    const float* __restrict__ b_out,  // [2]
    float* __restrict__ out)          // [256, 2]
{
    __shared__ float lds_w[HID * 2];  // staged W_out (6.4 KB)
    __shared__ float tile[16][16];

    const int lane  = threadIdx.x;        // 0..31
    const int mbase = blockIdx.x * 16;    // row-block of h
    const int nloc  = lane & 15;          // M index (A) / N index (B,C)
    const int khalf = (lane >> 4) * 2;    // 0 for lanes 0-15, 2 for 16-31

    // Stage W_out into LDS (unconditional, coalesced).
    for (int i = lane; i < HID * 2; i += 32) {
        lds_w[i] = W_out[i];
    }
    __syncthreads();

    // Zero-mask for padded columns; clamped index keeps every LDS read legal.
    const float wmask = (nloc < 2) ? 1.0f : 0.0f;
    const int   col   = nloc & 1;

    v8f c = {};
    const float* arow = hbuf + (size_t)(mbase + nloc) * HID;

    for (int k0 = 0; k0 < HID; k0 += 4) {
        const int ka = k0 + khalf;
        v2f a, b;
        a.x = arow[ka];
        a.y = arow[ka + 1];
        b.x = lds_w[ka * 2 + col]       * wmask;
        b.y = lds_w[(ka + 1) * 2 + col] * wmask;
        // 8 args: (neg_a, A, neg_b, B, c_mod, C, reuse_a, reuse_b)
        c = __builtin_amdgcn_wmma_f32_16x16x4_f32(
            false, a, false, b, (short)0, c, false, false);
    }

    // Scatter D tile to LDS per the C/D layout.
#pragma unroll
    for (int r = 0; r < 8; ++r) {
        int m = r + ((lane >> 4) * 8);
        tile[m][nloc] = c[r];
    }
    __syncthreads();

    // Per-row bias + 2-way softmax; 16 rows handled by lanes 0-15.
    if (lane < 16) {
        float o0 = tile[lane][0] + b_out[0];
        float o1 = tile[lane][1] + b_out[1];
        float mx = fmaxf(o0, o1);
        float e0 = expf(o0 - mx);
        float e1 = expf(o1 - mx);
        float inv = 1.0f / (e0 + e1);
        out[(size_t)(mbase + lane) * 2 + 0] = e0 * inv;
        out[(size_t)(mbase + lane) * 2 + 1] = e1 * inv;
    }
}

// ---------------------------------------------------------------------------
// Launch
// ---------------------------------------------------------------------------
extern "C" void kernel_launch(void* const* d_in, const int* in_sizes, int n_in,
                              void* d_out, int out_size, void* d_ws, size_t ws_size,
                              hipStream_t stream) {
    (void)in_sizes; (void)n_in; (void)out_size; (void)ws_size;

    const float* x     = (const float*)d_in[0];  // [64,4,1024,1,1] = [256,1024]
    const float* W_in  = (const float*)d_in[1];  // [1,800]
    const float* b_in  = (const float*)d_in[2];  // [800]
    const float* W_out = (const float*)d_in[3];  // [800,2]
    const float* b_out = (const float*)d_in[4];  // [2]
    float*       out   = (float*)d_out;          // [256,2]
    float*       hbuf  = (float*)d_ws;           // [256,800] = 800 KB

    rnn_scan_kernel<<<N_SEQ, 256, 0, stream>>>(x, W_in, b_in, hbuf);
    out_proj_softmax_kernel<<<N_SEQ / 16, 32, 0, stream>>>(hbuf, W_out, b_out, out);
}